// HierarchicalGNNClassifier_55121610277010
// MI455X (gfx1250) — compile-verified
//
#include <hip/hip_runtime.h>
#include <hip/hip_bf16.h>

typedef float v2f __attribute__((ext_vector_type(2)));
typedef float v8f __attribute__((ext_vector_type(8)));

#define HID 256
#define BATCH 2048

__device__ __forceinline__ v8f wmma_f32_4(v2f a, v2f b, v8f c) {
  // V_WMMA_F32_16X16X4_F32 : D(16x16 f32) = A(16x4) * B(4x16) + C
  return __builtin_amdgcn_wmma_f32_16x16x4_f32(false, a, false, b, (short)0, c,
                                               false, false);
}

__device__ __forceinline__ void async_b128_to_lds(unsigned lds_off,
                                                  const float* gaddr) {
  // GLOBAL_LOAD_ASYNC_TO_LDS_B128, GV mode (64-bit vaddr), ASYNCcnt tracked.
  asm volatile("global_load_async_to_lds_b128 %0, %1, off"
               :
               : "v"(lds_off), "v"(gaddr)
               : "memory");
}

// ---------------------------------------------------------------------------
// Tiled WMMA GEMM: C[M,N] = A[M,K] @ Bt[N,K]^T (+ bias[n]).
// Block = 256 threads (8 wave32), 128x128 C tile, K staged by 32 through LDS
// with double-buffered GLOBAL_LOAD_ASYNC_TO_LDS_B128.
// Both LDS tiles are [row][k] so every WMMA fragment is one ds_load_b64.
// ---------------------------------------------------------------------------
__global__ __launch_bounds__(256) void gemm_wmma(
    const float* __restrict__ A, const float* __restrict__ Bt,
    const float* __restrict__ bias, float* __restrict__ C, int M, int N, int K,
    int lda, int ldb, int ldc) {
  __shared__ float As[2][128 * 32];
  __shared__ float Bs[2][128 * 32];

  const int tid = threadIdx.x;
  const int m0 = blockIdx.y * 128;
  const int n0 = blockIdx.x * 128;
  const int wave = tid >> 5;
  const int lane = tid & 31;
  const int wm = (wave >> 2) * 64;  // 0 or 64
  const int wn = (wave & 3) * 32;   // 0,32,64,96
  const int hf = lane >> 4;         // half-wave select (K pair select)
  const int lrow = lane & 15;

  v8f acc[4][2];
#pragma unroll
  for (int i = 0; i < 4; ++i)
#pragma unroll
    for (int j = 0; j < 2; ++j) acc[i][j] = (v8f)0.0f;

  // Stage one 32-wide K slab into buffer `buf`. Exactly 8 async issues per
  // wave (deterministic ASYNCcnt); OOB rows/cols are clamped, never skipped —
  // the epilogue guards stores, so clamped garbage is never written out.
  auto stage = [&](int buf, int kb) {
#pragma unroll
    for (int uu = 0; uu < 4; ++uu) {
      int u = tid + uu * 256;  // 0..1023 float4 units
      int row = u >> 3;
      int kk = (u & 7) << 2;
      int gm = m0 + row;
      gm = gm < M ? gm : M - 1;
      async_b128_to_lds((unsigned)(uintptr_t)&As[buf][row * 32 + kk],
                        A + (size_t)gm * lda + kb + kk);
    }
#pragma unroll
    for (int uu = 0; uu < 4; ++uu) {
      int u = tid + uu * 256;
      int col = u >> 3;
      int kk = (u & 7) << 2;
      int gn = n0 + col;
      gn = gn < N ? gn : N - 1;
      async_b128_to_lds((unsigned)(uintptr_t)&Bs[buf][col * 32 + kk],
                        Bt + (size_t)gn * ldb + kb + kk);
    }
  };

  stage(0, 0);  // prologue

  for (int kb = 0, it = 0; kb < K; kb += 32, ++it) {
    const int cur = it & 1;
    const bool more = (kb + 32) < K;
    if (more) {
      stage(cur ^ 1, kb + 32);  // 8 newest in flight for next slab
      asm volatile("s_wait_asynccnt 0x8" ::: "memory");
    } else {
      asm volatile("s_wait_asynccnt 0x0" ::: "memory");
    }
    __syncthreads();  // current slab visible to all waves

#pragma unroll
    for (int k4 = 0; k4 < 8; ++k4) {
      const int kk = k4 * 4 + hf * 2;  // K pair for this half-wave
      v2f afr[4], bfr[2];
#pragma unroll
      for (int tm = 0; tm < 4; ++tm)
        afr[tm] = *(const v2f*)&As[cur][(wm + tm * 16 + lrow) * 32 + kk];
#pragma unroll
      for (int tn = 0; tn < 2; ++tn)
        bfr[tn] = *(const v2f*)&Bs[cur][(wn + tn * 16 + lrow) * 32 + kk];
#pragma unroll
      for (int tm = 0; tm < 4; ++tm)
#pragma unroll
        for (int tn = 0; tn < 2; ++tn)
          acc[tm][tn] = wmma_f32_4(afr[tm], bfr[tn], acc[tm][tn]);
    }
    __syncthreads();  // done reading `cur` before it is re-staged
  }

  // ---- epilogue: C layout = VGPR v: lanes0-15 M=v, lanes16-31 M=v+8 ----
#pragma unroll
  for (int tm = 0; tm < 4; ++tm) {
#pragma unroll
    for (int tn = 0; tn < 2; ++tn) {
      int gn = n0 + wn + tn * 16 + lrow;
      if (gn < N) {
        float bv = bias ? bias[gn] : 0.0f;
        int base_m = m0 + wm + tm * 16 + hf * 8;
#pragma unroll
        for (int v = 0; v < 8; ++v) {
          int gm = base_m + v;
          if (gm < M) C[(size_t)gm * ldc + gn] = acc[tm][tn][v] + bv;
        }
      }
    }
  }
}

// ---------------------------------------------------------------------------
// 32x32 LDS-tiled transpose: out[Ccols][R] = in[R][Ccols]^T  (small weights)
// ---------------------------------------------------------------------------
__global__ __launch_bounds__(256) void k_transpose(const float* __restrict__ in,
                                                   float* __restrict__ out,
                                                   int R, int Ccols) {
  __shared__ float t[32][33];
  int bx = blockIdx.x * 32, by = blockIdx.y * 32;
  int tx = threadIdx.x & 31, ty = threadIdx.x >> 5;
  for (int r = ty; r < 32; r += 8) {
    int gr = by + r, gc = bx + tx;
    t[r][tx] = (gr < R && gc < Ccols) ? in[(size_t)gr * Ccols + gc] : 0.f;
  }
  __syncthreads();
  for (int r = ty; r < 32; r += 8) {
    int gr = bx + r, gc = by + tx;
    if (gr < Ccols && gc < R) out[(size_t)gr * R + gc] = t[tx][r];
  }
}

// ---------------------------------------------------------------------------
// Degree / aggregation helpers (bandwidth trivial)
// ---------------------------------------------------------------------------
__device__ __forceinline__ void atomAddF(float* p, float v) {
  __hip_atomic_fetch_add(p, v, __ATOMIC_RELAXED, __HIP_MEMORY_SCOPE_AGENT);
}

__global__ void k_deg_init(float* deg, int n) {
  int i = blockIdx.x * blockDim.x + threadIdx.x;
  if (i < n) deg[i] = 1.0f;  // self loop
}

__global__ void k_deg_scatter(const int* __restrict__ dst, float* deg, int E) {
  int i = blockIdx.x * blockDim.x + threadIdx.x;
  if (i < E) atomAddF(&deg[dst[i]], 1.0f);
}

__global__ void k_deg_rsqrt(float* deg, int n) {
  int i = blockIdx.x * blockDim.x + threadIdx.x;
  if (i < n) deg[i] = rsqrtf(deg[i]);
}

// out[node,h] = h[node,h] * dinv[node]^2   (self-loop term)
__global__ void k_agg_init(const float* __restrict__ h,
                           const float* __restrict__ dinv,
                           float* __restrict__ out, int n) {
  size_t i = (size_t)blockIdx.x * blockDim.x + threadIdx.x;
  if (i < (size_t)n * HID) {
    int node = (int)(i >> 8);
    float d = dinv[node];
    out[i] = h[i] * d * d;
  }
}

// one edge per block (256 = HID threads), atomic scatter-add
__global__ void k_agg_scatter(const int* __restrict__ src,
                              const int* __restrict__ dst,
                              const float* __restrict__ dinv,
                              const float* __restrict__ h,
                              float* __restrict__ out, int E) {
  int e = blockIdx.x;
  if (e >= E) return;
  int s = src[e], d = dst[e];
  float w = dinv[s] * dinv[d];
  int hcol = threadIdx.x;
  atomAddF(&out[(size_t)d * HID + hcol], h[(size_t)s * HID + hcol] * w);
}

__global__ void k_bias_act(float* __restrict__ h, const float* __restrict__ b,
                           int n, int relu) {
  size_t i = (size_t)blockIdx.x * blockDim.x + threadIdx.x;
  if (i < (size_t)n * HID) {
    float v = h[i] + b[i & (HID - 1)];
    h[i] = relu ? fmaxf(v, 0.0f) : v;
  }
}

// ---------------------------------------------------------------------------
extern "C" void kernel_launch(void* const* d_in, const int* in_sizes, int n_in,
                              void* d_out, int out_size, void* d_ws,
                              size_t ws_size, hipStream_t stream) {
  const float* x = (const float*)d_in[0];      // [B, IN]
  const int* eidx = (const int*)d_in[1];       // [2, E]
  const float* emb = (const float*)d_in[2];    // [N, H]
  const float* W1 = (const float*)d_in[3];     // [H, H]
  const float* b1 = (const float*)d_in[4];     // [H]
  const float* W2 = (const float*)d_in[5];     // [H, H]
  const float* b2 = (const float*)d_in[6];     // [H]
  const float* fcW = (const float*)d_in[7];    // [IN, H]
  const float* fcb = (const float*)d_in[8];    // [H]
  const float* nbias = (const float*)d_in[9];  // [N]

  const int E = in_sizes[1] / 2;
  const int N = in_sizes[2] / HID;
  const int IN = in_sizes[0] / BATCH;
  const int* esrc = eidx;
  const int* edst = eidx + E;

  // scratch: hA lives at the front of d_out (dead before the final GEMM
  // overwrites it); everything else in d_ws.
  float* hA = (float*)d_out;
  float* ws = (float*)d_ws;
  float* dinv = ws;                          // N floats (padded region)
  float* hB = ws + 51200;                    // N*HID
  float* S = hB + (size_t)N * HID;           // BATCH*HID
  float* W1t = S + (size_t)BATCH * HID;      // HID*HID
  float* W2t = W1t + HID * HID;              // HID*HID
  float* fcWt = W2t + HID * HID;             // HID*IN

  const int nblkN = (N + 127) / 128;        // 391
  const int nElem = (N * HID + 255) / 256;  // node-feature element blocks

  // pre-transpose weights so every GEMM consumes Bt[N][K]
  k_transpose<<<dim3(HID / 32, HID / 32), 256, 0, stream>>>(W1, W1t, HID, HID);
  k_transpose<<<dim3(HID / 32, HID / 32), 256, 0, stream>>>(W2, W2t, HID, HID);
  k_transpose<<<dim3(HID / 32, IN / 32), 256, 0, stream>>>(fcW, fcWt, IN, HID);

  // degrees -> dinv
  k_deg_init<<<(N + 255) / 256, 256, 0, stream>>>(dinv, N);
  k_deg_scatter<<<(E + 255) / 256, 256, 0, stream>>>(edst, dinv, E);
  k_deg_rsqrt<<<(N + 255) / 256, 256, 0, stream>>>(dinv, N);

  // --- GCN layer 1: hA = emb @ W1 ---
  gemm_wmma<<<dim3(HID / 128, nblkN), 256, 0, stream>>>(
      emb, W1t, nullptr, hA, N, HID, HID, HID, HID, HID);
  k_agg_init<<<nElem, 256, 0, stream>>>(hA, dinv, hB, N);
  k_agg_scatter<<<E, HID, 0, stream>>>(esrc, edst, dinv, hA, hB, E);
  k_bias_act<<<nElem, 256, 0, stream>>>(hB, b1, N, 1);  // +b1, relu

  // --- GCN layer 2: hA = hB @ W2 ---
  gemm_wmma<<<dim3(HID / 128, nblkN), 256, 0, stream>>>(
      hB, W2t, nullptr, hA, N, HID, HID, HID, HID, HID);
  k_agg_init<<<nElem, 256, 0, stream>>>(hA, dinv, hB, N);
  k_agg_scatter<<<E, HID, 0, stream>>>(esrc, edst, dinv, hA, hB, E);
  k_bias_act<<<nElem, 256, 0, stream>>>(hB, b2, N, 0);  // +b2, no relu

  // --- sample_feat = x @ fc_W + fc_b ---
  gemm_wmma<<<dim3(HID / 128, BATCH / 128), 256, 0, stream>>>(
      x, fcWt, fcb, S, BATCH, HID, IN, IN, IN, HID);

  // --- logits = S @ nf^T + node_bias  (dominant GEMM, writes d_out) ---
  gemm_wmma<<<dim3(nblkN, BATCH / 128), 256, 0, stream>>>(
      S, hB, nbias, (float*)d_out, BATCH, N, HID, HID, HID, N);
}